// CstLoss_90649579749529
// MI455X (gfx1250) — compile-verified
//
#include <hip/hip_runtime.h>
#include <hip/hip_bf16.h>
#include <math.h>

typedef __attribute__((ext_vector_type(2))) float v2f;
typedef __attribute__((ext_vector_type(8))) float v8f;

#define N_B  64
#define C_C  17
#define HW   128
#define TILE (HW * HW)
#define NC   (N_B * C_C)                 // 1088 tiles per tensor
#define CPAD 32                          // channels padded to 2 WMMA tiles
#define N_STRIDE   (CPAD * HW)           // 4096 floats per (proj,t,n) slab
#define T_STRIDE   (N_B * N_STRIDE)      // 262144
#define PROJ_STRIDE (2 * T_STRIDE)       // 524288
#define ACC_OFF    (2 * PROJ_STRIDE)     // 1048576: then {sum_all, sum_diag}
#define WS_FLOATS  (ACC_OFF + 2)

__device__ __forceinline__ float wave_sum(float v) {
#pragma unroll
    for (int off = 16; off > 0; off >>= 1) v += __shfl_xor(v, off, 32);
    return v;
}

// Zero the padded workspace + accumulators each call (deterministic, capture-safe).
__global__ __launch_bounds__(256) void zero_ws_kernel(float* __restrict__ ws, int count) {
    int i = blockIdx.x * 256 + threadIdx.x;
    if (i < count) ws[i] = 0.0f;
}

// One block per (tensor, n, c) 128x128 tile: row/col mean reduction fused with
// softmax + l2-normalization of both marginal vectors. Writes into the padded
// [proj][t][n][32][128] workspace (rows 17..31 stay zero).
__global__ __launch_bounds__(256) void marginals_softmax_kernel(
    const float* __restrict__ outp, const float* __restrict__ tgt,
    float* __restrict__ ws)
{
    __shared__ float rowsum[HW];
    __shared__ float colsum[HW];
    __shared__ float red[256];

    const int bid = blockIdx.x;              // 0 .. 2*NC-1
    const int t   = (bid >= NC) ? 1 : 0;     // 0 = output, 1 = target
    const int idx = bid - t * NC;            // n*17 + c
    const float* src = (t ? tgt : outp) + (size_t)idx * TILE;

    const int tid  = threadIdx.x;
    const int lane = tid & 31;
    const int wid  = tid >> 5;

    if (tid < HW) { rowsum[tid] = 0.0f; colsum[tid] = 0.0f; }
    __syncthreads();

    // Phase B: wave `wid` owns rows wid+8i; lane owns a float4 column slice.
    const float4* src4 = (const float4*)src;
    float4 cacc = make_float4(0.f, 0.f, 0.f, 0.f);
#pragma unroll
    for (int i = 0; i < 16; ++i) {
        const int r = wid + (i << 3);
        const float4 v = src4[r * 32 + lane];       // 512B coalesced per wave
        cacc.x += v.x; cacc.y += v.y; cacc.z += v.z; cacc.w += v.w;
        const float rs = wave_sum(v.x + v.y + v.z + v.w);
        if (lane == 0) rowsum[r] = rs;              // row fully owned by this wave
    }
    atomicAdd(&colsum[lane * 4 + 0], cacc.x);       // ds_add_f32
    atomicAdd(&colsum[lane * 4 + 1], cacc.y);
    atomicAdd(&colsum[lane * 4 + 2], cacc.z);
    atomicAdd(&colsum[lane * 4 + 3], cacc.w);
    __syncthreads();

    // Phase C: threads [0,128) -> h-projection (rowsum), [128,256) -> w-projection.
    const int proj  = tid >> 7;
    const int e     = tid & 127;
    const int rbase = proj << 7;
    const float x = (proj ? colsum[e] : rowsum[e]) * (1.0f / 128.0f);  // mean

    // max
    red[tid] = x; __syncthreads();
#pragma unroll
    for (int s = 64; s > 0; s >>= 1) {
        if (e < s) red[tid] = fmaxf(red[tid], red[tid + s]);
        __syncthreads();
    }
    const float mx = red[rbase]; __syncthreads();

    // sum of exp
    const float ex = expf(x - mx);
    red[tid] = ex; __syncthreads();
#pragma unroll
    for (int s = 64; s > 0; s >>= 1) {
        if (e < s) red[tid] += red[tid + s];
        __syncthreads();
    }
    const float ssum = red[rbase]; __syncthreads();
    const float p = ex / ssum;

    // sum of squares for l2 norm
    red[tid] = p * p; __syncthreads();
#pragma unroll
    for (int s = 64; s > 0; s >>= 1) {
        if (e < s) red[tid] += red[tid + s];
        __syncthreads();
    }
    const float q = red[rbase];

    const float nrm  = fmaxf(sqrtf(q), 1e-8f);      // torch-style eps clamp
    const float outv = p / nrm;

    const int nn = idx / C_C;
    const int cc = idx - nn * C_C;
    float* dst = ws + (size_t)proj * PROJ_STRIDE + (size_t)t * T_STRIDE
                    + (size_t)nn * N_STRIDE + (size_t)cc * HW + e;
    *dst = outv;
}

// One block per (n, projection); 4 waves = 2x2 grid of 16x16 WMMA tiles over
// the padded 32x32 output of A(32x128) * B(32x128)^T (rows 17..31 are zero),
// K swept 4 at a time with V_WMMA_F32_16X16X4_F32 (exact f32 math).
// Padded layout => unconditional b64 fragment loads, no exec-mask churn.
__global__ __launch_bounds__(128) void wmma_pairs_kernel(
    const float* __restrict__ ws, float* __restrict__ acc_out)
{
    const int n    = blockIdx.x & 63;
    const int proj = blockIdx.x >> 6;
    const float* base = ws + (size_t)proj * PROJ_STRIDE + (size_t)n * N_STRIDE;
    const float* A = base;               // output (t=0)
    const float* B = base + T_STRIDE;    // target (t=1)

    const int tid  = threadIdx.x;
    const int lane = tid & 31;
    const int wid  = tid >> 5;
    const int tm   = wid >> 1;      // M tile
    const int tn   = wid & 1;       // N tile
    const int half = lane >> 4;     // lanes 16-31 carry K+2,K+3
    const int l15  = lane & 15;

    const int arow = tm * 16 + l15;  // 0..31, always valid in padded layout
    const int bcol = tn * 16 + l15;
    const float* Ap = A + arow * HW + half * 2;
    const float* Bp = B + bcol * HW + half * 2;

    v8f acc = {0.f, 0.f, 0.f, 0.f, 0.f, 0.f, 0.f, 0.f};
#pragma unroll
    for (int k = 0; k < HW; k += 4) {
        const v2f a = *(const v2f*)(Ap + k);   // global_load_b64, imm offset
        const v2f b = *(const v2f*)(Bp + k);
        // 8 args: (neg_a, A, neg_b, B, c_mod, C, reuse_a, reuse_b)
        acc = __builtin_amdgcn_wmma_f32_16x16x4_f32(
            false, a, false, b, (short)0, acc, false, false);
    }

    // C layout: VGPR r -> M = r + 8*half, N = l15
    float lall = 0.f, ldiag = 0.f;
#pragma unroll
    for (int r = 0; r < 8; ++r) {
        const int i = tm * 16 + r + (half << 3);
        const int j = tn * 16 + l15;
        if (i < C_C && j < C_C) {
            lall += acc[r];
            if (i == j) ldiag += acc[r];
        }
    }
    lall  = wave_sum(lall);
    ldiag = wave_sum(ldiag);
    if (lane == 0) {
        atomicAdd(&acc_out[0], lall);    // global_atomic_add_f32
        atomicAdd(&acc_out[1], ldiag);
    }
}

__global__ void finalize_kernel(const float* __restrict__ acc, float* __restrict__ out) {
    const float all  = acc[0];
    const float diag = acc[1];
    const float sim_pos = 0.5f * diag / (float)(N_B * C_C);
    const float sim     = 0.5f * all  / (float)N_B;
    out[0] = -logf(sim_pos / sim) / (float)C_C / (float)N_B;
}

extern "C" void kernel_launch(void* const* d_in, const int* in_sizes, int n_in,
                              void* d_out, int out_size, void* d_ws, size_t ws_size,
                              hipStream_t stream) {
    (void)in_sizes; (void)n_in; (void)out_size; (void)ws_size;
    const float* outp = (const float*)d_in[0];
    const float* tgt  = (const float*)d_in[1];
    float* ws  = (float*)d_ws;
    float* acc = ws + ACC_OFF;

    zero_ws_kernel<<<(WS_FLOATS + 255) / 256, 256, 0, stream>>>(ws, WS_FLOATS);
    marginals_softmax_kernel<<<2 * NC, 256, 0, stream>>>(outp, tgt, ws);
    wmma_pairs_kernel<<<2 * N_B, 128, 0, stream>>>(ws, acc);
    finalize_kernel<<<1, 1, 0, stream>>>(acc, (float*)d_out);
}